// LSTM_41575283425452
// MI455X (gfx1250) — compile-verified
//
#include <hip/hip_runtime.h>

// LSTM: T=4096, B=1024, I=5, H=8, O=5  (float32 throughout)
// Kernel 1: fused input-projection + recurrent scan. One wave per batch
//           element, one gate-lane per lane (4H == 32 == wave32).
// Kernel 2: output projection via V_WMMA_F32_16X16X4_F32 (K=8 -> 2 chained).

#define T_STEPS 4096
#define BATCH   1024
#define IN_DIM  5
#define HID     8
#define OUT_DIM 5

typedef __attribute__((ext_vector_type(2))) float v2f;
typedef __attribute__((ext_vector_type(8))) float v8f;

__device__ __forceinline__ float bcast_lane(float v, int l) {
  return __int_as_float(__builtin_amdgcn_readlane(__float_as_int(v), l));
}

__global__ __launch_bounds__(256) void lstm_scan_kernel(
    const float* __restrict__ seq, const float* __restrict__ W_ih,
    const float* __restrict__ W_hh, const float* __restrict__ b_ih,
    const float* __restrict__ b_hh, float* __restrict__ h_out) {
  const int lane  = threadIdx.x & 31;
  const int wave  = threadIdx.x >> 5;
  const int b     = blockIdx.x * 8 + wave;   // batch element for this wave
  const int g     = lane;                    // gate index 0..31 (i,f,g,o x8)
  const int gtype = lane >> 3;               // 0=i 1=f 2=g(tanh) 3=o

  const float NEG_L2E = -1.44269504088896340736f;
  // Fold the tanh pre-scale (z = 2x for tanh lanes) and -log2(e) into the
  // weights so the hot loop computes acc = -z*log2e directly:
  //   sigmoid(z) = rcp(1 + exp2(acc));  act = sa*sigmoid + sb
  const float wscale = (gtype == 2) ? (2.0f * NEG_L2E) : NEG_L2E;
  const float sa     = (gtype == 2) ? 2.0f : 1.0f;
  const float sb     = (gtype == 2) ? -1.0f : 0.0f;

  float wih[IN_DIM];
#pragma unroll
  for (int i = 0; i < IN_DIM; ++i) wih[i] = W_ih[g * IN_DIM + i] * wscale;
  float whh[HID];
#pragma unroll
  for (int k = 0; k < HID; ++k) whh[k] = W_hh[g * HID + k] * wscale;
  const float bias = (b_ih[g] + b_hh[g]) * wscale;

  float h = 0.0f, c = 0.0f;            // state lives on lanes 0..7 (unit = lane)
  const float* xp = seq + (size_t)b * IN_DIM;      // x[t=0, b, :]
  const int tstride = BATCH * IN_DIM;              // elements per timestep

  for (int t = 0; t < T_STEPS; ++t) {
    // Stay ahead of HBM latency: each step touches a new 20KB-strided line.
    __builtin_prefetch(xp + 16 * tstride, 0, 0);   // global_prefetch_b8

    const float x0 = xp[0], x1 = xp[1], x2 = xp[2], x3 = xp[3], x4 = xp[4];

    // Gate pre-activation (already scaled): two accumulators to shorten chain.
    float a0 = __builtin_fmaf(x0, wih[0], bias);
    float a1 = x2 * wih[2];
    a0 = __builtin_fmaf(x1, wih[1], a0);
    a1 = __builtin_fmaf(x3, wih[3], a1);
    a0 = __builtin_fmaf(x4, wih[4], a0);
#pragma unroll
    for (int k = 0; k < HID; k += 2) {
      a0 = __builtin_fmaf(bcast_lane(h, k),     whh[k],     a0);  // v_fmac s,v
      a1 = __builtin_fmaf(bcast_lane(h, k + 1), whh[k + 1], a1);
    }
    const float z   = a0 + a1;                          // = -pre*log2e (x2 tanh)
    const float e   = __builtin_amdgcn_exp2f(z);        // v_exp_f32
    const float s   = __builtin_amdgcn_rcpf(1.0f + e);  // sigmoid(pre)
    const float act = __builtin_fmaf(sa, s, sb);        // tanh lanes: 2s-1

    // Gather f,g,o into lanes 0..7 (which hold i and the cell state).
    const float fv = __shfl(act, lane + 8,  32);
    const float gv = __shfl(act, lane + 16, 32);
    const float ov = __shfl(act, lane + 24, 32);

    c = __builtin_fmaf(act, gv, fv * c);                // c = f*c + i*g
    const float e2 = __builtin_amdgcn_exp2f(c * (2.0f * NEG_L2E));
    const float th = __builtin_fmaf(2.0f, __builtin_amdgcn_rcpf(1.0f + e2), -1.0f);
    h = ov * th;                                        // valid on lanes 0..7

    xp += tstride;
  }
  if (lane < HID) h_out[(size_t)b * HID + lane] = h;
}

// out(1024x5) = h_last(1024x8) @ W_out^T(8x5) + b_out, via WMMA f32 16x16x4.
// One wave per 16-batch tile. A = h tile (16x8), B = W_out^T zero-padded to
// 8x16 columns, C seeded with broadcast bias. K=8 -> two chained WMMAs.
__global__ __launch_bounds__(128) void out_proj_wmma(
    const float* __restrict__ h_last, const float* __restrict__ W_out,
    const float* __restrict__ b_out, float* __restrict__ out) {
  const int lane = threadIdx.x & 31;
  const int tile = (blockIdx.x * blockDim.x + threadIdx.x) >> 5;  // 0..63
  const int half = lane >> 4;    // lane half selects K/M sub-block per ISA layout
  const int lm   = lane & 15;    // A: row M ; B/C/D: column N

  // A 16x4 f32 layout: VGPR0 -> K = 2*half, VGPR1 -> K = 2*half+1.
  const float* hrow = h_last + (size_t)(tile * 16 + lm) * HID;
  v2f a_lo, a_hi;
  a_lo.x = hrow[0 + 2 * half];
  a_lo.y = hrow[1 + 2 * half];
  a_hi.x = hrow[4 + 2 * half];
  a_hi.y = hrow[5 + 2 * half];

  // B 4x16 f32 (mirrored layout): element (K, N=lm), zero for N >= OUT_DIM.
  float w0 = 0.f, w1 = 0.f, w2 = 0.f, w3 = 0.f, bo = 0.f;
  if (lm < OUT_DIM) {
    const float* wr = W_out + lm * HID;   // W_out^T[k][n] = W_out[n][k]
    w0 = wr[0 + 2 * half];
    w1 = wr[1 + 2 * half];
    w2 = wr[4 + 2 * half];
    w3 = wr[5 + 2 * half];
    bo = b_out[lm];
  }
  v2f b_lo, b_hi;
  b_lo.x = w0; b_lo.y = w1;
  b_hi.x = w2; b_hi.y = w3;

  v8f acc;
#pragma unroll
  for (int r = 0; r < 8; ++r) acc[r] = bo;   // C = bias broadcast per column

  acc = __builtin_amdgcn_wmma_f32_16x16x4_f32(false, a_lo, false, b_lo,
                                              (short)0, acc, false, false);
  acc = __builtin_amdgcn_wmma_f32_16x16x4_f32(false, a_hi, false, b_hi,
                                              (short)0, acc, false, false);

  // D layout: VGPR r holds row M = r + 8*half, column N = lm.
  if (lm < OUT_DIM) {
    float* orow = out + (size_t)(tile * 16 + 8 * half) * OUT_DIM + lm;
#pragma unroll
    for (int r = 0; r < 8; ++r) orow[r * OUT_DIM] = acc[r];
  }
}

extern "C" void kernel_launch(void* const* d_in, const int* in_sizes, int n_in,
                              void* d_out, int out_size, void* d_ws, size_t ws_size,
                              hipStream_t stream) {
  const float* seq   = (const float*)d_in[0];
  const float* W_ih  = (const float*)d_in[1];
  const float* W_hh  = (const float*)d_in[2];
  const float* b_ih  = (const float*)d_in[3];
  const float* b_hh  = (const float*)d_in[4];
  const float* W_out = (const float*)d_in[5];
  const float* b_out = (const float*)d_in[6];
  float* out    = (float*)d_out;
  float* h_last = (float*)d_ws;   // 1024*8 floats = 32 KB of scratch

  // 1024 waves total: 8 waves (8 batch elements) per 256-thread block.
  lstm_scan_kernel<<<BATCH / 8, 256, 0, stream>>>(seq, W_ih, W_hh, b_ih, b_hh,
                                                  h_last);
  // 64 tiles of 16 batch rows, 4 waves per 128-thread block -> 16 blocks.
  out_proj_wmma<<<(BATCH / 16) * 32 / 128, 128, 0, stream>>>(h_last, W_out,
                                                             b_out, out);
}